// OTFL_6141803233289
// MI455X (gfx1250) — compile-verified
//
#include <hip/hip_runtime.h>
#include <hip/hip_bf16.h>
#include <math.h>
#include <stdint.h>

// Problem constants (from reference setup_inputs)
#define BATCH   8192
#define NDIM    2048
#define NCLS    100
#define CP      128       // NCLS padded to 8 WMMA n-tiles (one per wave)
#define NTILES  8         // CP/16
#define BK      32        // K-chunk staged in LDS

// LDS row strides (padded for bank behavior; TDM applies these pads itself)
#define A_STRIDE 36       // 32 + 4
#define B_STRIDE 136      // 128 + 8  (2*136 % 64 == 16 -> lane halves on disjoint banks)

typedef float v2f __attribute__((ext_vector_type(2)));
typedef float v8f __attribute__((ext_vector_type(8)));
typedef unsigned v4u __attribute__((ext_vector_type(4)));
typedef int v4i_t __attribute__((ext_vector_type(4)));
typedef int v8i_t __attribute__((ext_vector_type(8)));

#if defined(__has_builtin)
#if __has_builtin(__builtin_amdgcn_tensor_load_to_lds) && __has_builtin(__builtin_amdgcn_s_wait_tensorcnt)
#define USE_TDM 1
#endif
#endif
#ifndef USE_TDM
#define USE_TDM 0
#endif

#if __has_include(<hip/amd_detail/amd_gfx1250_TDM.h>)
#define TDM_SIX_ARGS 1
#else
#define TDM_SIX_ARGS 0
#endif

// ---------------------------------------------------------------------------
// 0) zero the accumulators (Pacc + counts + sums)
// ---------------------------------------------------------------------------
__global__ void zero_kernel(float* p, int n) {
    int i = blockIdx.x * blockDim.x + threadIdx.x;
    if (i < n) p[i] = 0.0f;
}

// ---------------------------------------------------------------------------
// 1) Wp[k, c] = (c < NCLS) ? W[k, c] : 0   (2048 x 128)
// ---------------------------------------------------------------------------
__global__ void pad_w_kernel(const float* __restrict__ W, float* __restrict__ Wp) {
    int i = blockIdx.x * blockDim.x + threadIdx.x;
    if (i >= NDIM * CP) return;
    int k = i / CP, c = i % CP;
    Wp[i] = (c < NCLS) ? W[k * NCLS + c] : 0.0f;
}

// ---------------------------------------------------------------------------
// Generic f32 WMMA GEMM (guarded, strided A) — used for the two tiny GEMMs
// and as fallback for px.  blockDim = 256 (8 waves, one 16-wide n-tile each).
// V_WMMA_F32_16X16X4_F32 lane layouts per ISA §7.12.2.
// ---------------------------------------------------------------------------
__global__ __launch_bounds__(256) void gemm_wmma_kernel(
    const float* __restrict__ A, const float* __restrict__ Bm,
    float* __restrict__ C, int M, int K, long a_rs, long a_cs)
{
    __shared__ float lds_a[16 * BK];          // [m][k]
    __shared__ float lds_b[CP * (BK + 1)];    // [n][k], padded stride

    const int tid   = threadIdx.x;
    const int wave  = tid >> 5;
    const int lane  = tid & 31;
    const int m0    = blockIdx.x * 16;
    const int n0    = wave * 16;
    const int lhalf = lane >> 4;
    const int l16   = lane & 15;

    v8f acc = {};

    for (int k0 = 0; k0 < K; k0 += BK) {
        for (int idx = tid; idx < 16 * BK; idx += 256) {
            int r = idx / BK, c = idx % BK;
            int row = m0 + r;
            float v = 0.0f;
            if (row < M) v = A[(long)row * a_rs + (long)(k0 + c) * a_cs];
            lds_a[r * BK + c] = v;
        }
        for (int idx = tid; idx < BK * CP; idx += 256) {
            int r = idx / CP, c = idx % CP;                 // coalesced read
            lds_b[c * (BK + 1) + r] = Bm[(long)(k0 + r) * CP + c];
        }
        __syncthreads();

        #pragma unroll
        for (int kk = 0; kk < BK; kk += 4) {
            int ak = kk + 2 * lhalf;
            v2f a, b;
            a.x = lds_a[l16 * BK + ak];
            a.y = lds_a[l16 * BK + ak + 1];
            b.x = lds_b[(n0 + l16) * (BK + 1) + ak];
            b.y = lds_b[(n0 + l16) * (BK + 1) + ak + 1];
            acc = __builtin_amdgcn_wmma_f32_16x16x4_f32(
                false, a, false, b, (short)0, acc, false, false);
        }
        __syncthreads();
    }

    #pragma unroll
    for (int v = 0; v < 8; ++v) {
        int row = m0 + v + 8 * lhalf;
        if (row < M) C[(long)row * CP + n0 + l16] = acc[v];
    }
}

#if USE_TDM
// ---------------------------------------------------------------------------
// TDM 2D tile load: builds the D# (groups 0/1 per ISA §8.3/§8.4) and issues
// TENSOR_LOAD_TO_LDS.  tensor dims == tile dims (tiles always in-bounds).
//   pad_enable applies LDS row padding: interval 2^(ic+1) dwords, amount ac+1.
// ---------------------------------------------------------------------------
__device__ __forceinline__ void tdm_load_2d(
    unsigned lds_addr, unsigned long long gaddr,
    unsigned tile0, unsigned tile1, unsigned stride0,
    unsigned pad_interval_code, unsigned pad_amount_code)
{
    v4u g0;
    g0.x = 1u;                                          // count=1, user mode
    g0.y = lds_addr;                                    // lds_addr[31:0]
    g0.z = (unsigned)(gaddr & 0xffffffffu);             // global_addr[31:0]
    g0.w = (unsigned)((gaddr >> 32) & 0x01ffffffu) | (2u << 30);  // addr[56:32] | type=2

    // group1: data_size=4B(2), pad_enable, pad codes; tensor dims == tile dims
    unsigned d0 = (2u << 16) | (1u << 20) | (pad_interval_code << 22) | (pad_amount_code << 25);
    v8i_t g1;
    g1[0] = (int)d0;
    g1[1] = (int)((tile0 & 0xffffu) << 16);             // tensor_dim0[15:0]
    g1[2] = (int)((tile0 >> 16) | ((tile1 & 0xffffu) << 16)); // td0 hi | td1 lo
    g1[3] = (int)((tile1 >> 16) | ((tile0 & 0xffffu) << 16)); // td1 hi | tile_dim0
    g1[4] = (int)(tile1 & 0xffffu);                     // tile_dim1 (tile_dim2 = 0)
    g1[5] = (int)stride0;                               // tensor_dim0_stride[31:0]
    g1[6] = 0;
    g1[7] = 0;

    v4i_t z4 = {0, 0, 0, 0};
#if TDM_SIX_ARGS
    v8i_t z8 = {0, 0, 0, 0, 0, 0, 0, 0};
    __builtin_amdgcn_tensor_load_to_lds(g0, g1, z4, z4, z8, 0);
#else
    __builtin_amdgcn_tensor_load_to_lds(g0, g1, z4, z4, 0);
#endif
}

// ---------------------------------------------------------------------------
// 2) px = x @ Wp, TDM-fed, double-buffered LDS.  grid = BATCH/16, block = 256.
// ---------------------------------------------------------------------------
__global__ __launch_bounds__(256) void gemm_px_tdm_kernel(
    const float* __restrict__ A, const float* __restrict__ Bm, float* __restrict__ C)
{
    __shared__ float ldsA[2][16 * A_STRIDE];
    __shared__ float ldsB[2][BK * B_STRIDE];

    const int tid   = threadIdx.x;
    const int wave  = tid >> 5;
    const int lane  = tid & 31;
    const int m0    = blockIdx.x * 16;
    const int n0    = wave * 16;
    const int lhalf = lane >> 4;
    const int l16   = lane & 15;

    const unsigned aBase[2] = { (unsigned)(uintptr_t)&ldsA[0][0], (unsigned)(uintptr_t)&ldsA[1][0] };
    const unsigned bBase[2] = { (unsigned)(uintptr_t)&ldsB[0][0], (unsigned)(uintptr_t)&ldsB[1][0] };
    const unsigned long long ga = (unsigned long long)(uintptr_t)(A + (long)m0 * NDIM);
    const unsigned long long gb = (unsigned long long)(uintptr_t)Bm;

    // prologue: tile 0 into buffer 0
    if (tid == 0) {
        tdm_load_2d(aBase[0], ga, BK, 16, NDIM, 4u, 3u);        // 16x32, row 32->36 dw
        tdm_load_2d(bBase[0], gb, CP, BK, CP, 6u, 7u);          // 32x128, row 128->136 dw
    }

    v8f acc = {};
    const int NT = NDIM / BK;
    for (int kt = 0; kt < NT; ++kt) {
        const int cur = kt & 1;
        if (tid == 0) {
            if (kt + 1 < NT) {
                const unsigned long long koff = (unsigned long long)(kt + 1) * BK * 4ull;
                tdm_load_2d(aBase[1 - cur], ga + koff, BK, 16, NDIM, 4u, 3u);
                tdm_load_2d(bBase[1 - cur], gb + koff * CP, CP, BK, CP, 6u, 7u);
                __builtin_amdgcn_s_wait_tensorcnt(2);   // current tile's 2 DMAs done
            } else {
                __builtin_amdgcn_s_wait_tensorcnt(0);
            }
        }
        __syncthreads();                                // data visible to all waves

        const float* la = &ldsA[cur][0];
        const float* lb = &ldsB[cur][0];
        #pragma unroll
        for (int kk = 0; kk < BK; kk += 4) {
            int ak = kk + 2 * lhalf;
            v2f a, b;
            a.x = la[l16 * A_STRIDE + ak];
            a.y = la[l16 * A_STRIDE + ak + 1];
            b.x = lb[ak * B_STRIDE + n0 + l16];
            b.y = lb[(ak + 1) * B_STRIDE + n0 + l16];
            acc = __builtin_amdgcn_wmma_f32_16x16x4_f32(
                false, a, false, b, (short)0, acc, false, false);
        }
        __syncthreads();                                // all reads done before overwrite
    }

    #pragma unroll
    for (int v = 0; v < 8; ++v) {
        int row = m0 + v + 8 * lhalf;
        C[(long)row * CP + n0 + l16] = acc[v];
    }
}
#endif  // USE_TDM

// ---------------------------------------------------------------------------
// 3) V[c, :] = softmax(pa[c, :100]) - e_c   (one wave per row)
// ---------------------------------------------------------------------------
__global__ __launch_bounds__(256) void pa_to_v_kernel(const float* __restrict__ pa,
                                                      float* __restrict__ V)
{
    int wave = threadIdx.x >> 5, lane = threadIdx.x & 31;
    int row = blockIdx.x * 8 + wave;
    if (row >= NCLS) return;
    const float* p = pa + (long)row * CP;
    float vals[4];
    float mx = -1e30f;
    #pragma unroll
    for (int j = 0; j < 4; ++j) {
        int c = lane + 32 * j;
        vals[j] = (c < NCLS) ? p[c] : -1e30f;
        mx = fmaxf(mx, vals[j]);
    }
    for (int off = 16; off; off >>= 1) mx = fmaxf(mx, __shfl_xor(mx, off, 32));
    float se = 0.0f;
    #pragma unroll
    for (int j = 0; j < 4; ++j) {
        int c = lane + 32 * j;
        vals[j] = (c < NCLS) ? __expf(vals[j] - mx) : 0.0f;
        se += vals[j];
    }
    for (int off = 16; off; off >>= 1) se += __shfl_xor(se, off, 32);
    float inv = 1.0f / se;
    float* v = V + (long)row * CP;
    #pragma unroll
    for (int j = 0; j < 4; ++j) {
        int c = lane + 32 * j;
        if (c < CP) v[c] = (c < NCLS) ? (vals[j] * inv - (c == row ? 1.0f : 0.0f)) : 0.0f;
    }
}

// ---------------------------------------------------------------------------
// 4) U2[c, m] = sum_k V[c,k] * G[k,m]   (tiny 100x100x100)
// ---------------------------------------------------------------------------
__global__ void u2_kernel(const float* __restrict__ V, const float* __restrict__ G,
                          float* __restrict__ U2)
{
    int c = blockIdx.x;           // 0..NCLS-1
    int m = threadIdx.x;          // 0..127
    if (m >= CP) return;
    float s = 0.0f;
    if (m < NCLS)
        for (int k = 0; k < NCLS; ++k) s += V[c * CP + k] * G[k * CP + m];
    U2[c * CP + m] = s;
}

// ---------------------------------------------------------------------------
// 5) Per-sample: softmax of px row, ce, pos; accumulate Pacc/counts/sums
// ---------------------------------------------------------------------------
__global__ __launch_bounds__(256) void sample_kernel(
    const float* __restrict__ px, const int* __restrict__ y,
    const float* __restrict__ U2, float* __restrict__ Pacc,
    float* __restrict__ counts, float* __restrict__ sums)
{
    __shared__ float red[16];
    int wave = threadIdx.x >> 5, lane = threadIdx.x & 31;
    long i = (long)blockIdx.x * 8 + wave;
    float ce = 0.0f, pos = 0.0f;
    if (i < BATCH) {
        const float* p = px + i * CP;
        int yi = y[i];
        float vals[4];
        float mx = -1e30f;
        #pragma unroll
        for (int j = 0; j < 4; ++j) {
            int c = lane + 32 * j;
            vals[j] = (c < NCLS) ? p[c] : -1e30f;
            mx = fmaxf(mx, vals[j]);
        }
        for (int off = 16; off; off >>= 1) mx = fmaxf(mx, __shfl_xor(mx, off, 32));
        float se = 0.0f;
        #pragma unroll
        for (int j = 0; j < 4; ++j) {
            int c = lane + 32 * j;
            vals[j] = (c < NCLS) ? __expf(vals[j] - mx) : 0.0f;
            se += vals[j];
        }
        for (int off = 16; off; off >>= 1) se += __shfl_xor(se, off, 32);
        float inv = 1.0f / se;
        const float* u = U2 + (long)yi * CP;
        float dot = 0.0f;
        #pragma unroll
        for (int j = 0; j < 4; ++j) {
            int c = lane + 32 * j;
            if (c < NCLS) {
                float s = vals[j] * inv;
                dot += s * u[c];
                atomicAdd(&Pacc[yi * CP + c], s);
            }
        }
        for (int off = 16; off; off >>= 1) dot += __shfl_xor(dot, off, 32);
        float pyi = p[yi];
        ce  = __logf(se) + mx - pyi;
        pos = dot - u[yi];
        if (lane == 0) atomicAdd(&counts[yi], 1.0f);
    }
    if (lane == 0) { red[wave] = ce; red[8 + wave] = pos; }
    __syncthreads();
    if (threadIdx.x == 0) {
        float a = 0.0f, b = 0.0f;
        for (int w = 0; w < 8; ++w) { a += red[w]; b += red[8 + w]; }
        atomicAdd(&sums[0], a);
        atomicAdd(&sums[1], b);
    }
}

// ---------------------------------------------------------------------------
// 6) Final: T, P_c, neg_sum = sum_c n_c (T - P_c) . U2[c];  write loss
// ---------------------------------------------------------------------------
__global__ void final_kernel(const float* __restrict__ Pacc, const float* __restrict__ counts,
                             const float* __restrict__ U2, const float* __restrict__ sums,
                             float* __restrict__ out)
{
    __shared__ float T[CP];
    __shared__ float red[128];
    int t = threadIdx.x;              // 128 threads
    if (t < CP) {
        float s = 0.0f;
        if (t < NCLS) {
            for (int c = 0; c < NCLS; ++c) s += Pacc[c * CP + t];
            s -= counts[t];
        }
        T[t] = s;
    }
    __syncthreads();
    float acc = 0.0f;
    if (t < NCLS) {
        int c = t;
        float nc = counts[c];
        float s = 0.0f;
        for (int m = 0; m < NCLS; ++m) {
            float Pm = Pacc[c * CP + m] - ((m == c) ? nc : 0.0f);
            s += (T[m] - Pm) * U2[c * CP + m];
        }
        acc = nc * s;
    }
    red[t] = acc;
    __syncthreads();
    for (int off = 64; off; off >>= 1) {
        if (t < off) red[t] += red[t + off];
        __syncthreads();
    }
    if (t == 0) {
        const float invB = 1.0f / (float)BATCH;
        float loss = sums[0] * invB + 2.0f * ((red[0] - sums[1]) * invB + 0.05f);
        out[0] = loss;
    }
}

// ---------------------------------------------------------------------------
// launch
// ---------------------------------------------------------------------------
extern "C" void kernel_launch(void* const* d_in, const int* in_sizes, int n_in,
                              void* d_out, int out_size, void* d_ws, size_t ws_size,
                              hipStream_t stream)
{
    const float* x       = (const float*)d_in[0];   // (B, 1, D)
    const float* W       = (const float*)d_in[1];   // (D, C)
    const float* anchors = (const float*)d_in[2];   // (C, 1, D)
    const int*   y       = (const int*)d_in[3];     // (B,)
    float* out = (float*)d_out;

    float* ws = (float*)d_ws;
    float* Wp     = ws;                                  // D*CP
    float* px     = Wp     + (size_t)NDIM * CP;          // B*CP
    float* pa     = px     + (size_t)BATCH * CP;         // CP*CP
    float* G      = pa     + (size_t)CP * CP;            // CP*CP
    float* V      = G      + (size_t)CP * CP;            // CP*CP
    float* U2     = V      + (size_t)CP * CP;            // CP*CP
    float* Pacc   = U2     + (size_t)CP * CP;            // NCLS x CP
    float* counts = Pacc   + (size_t)NCLS * CP;          // CP
    float* sums   = counts + CP;                         // [0]=ce, [1]=pos

    int nzero = NCLS * CP + CP + 2;
    zero_kernel<<<(nzero + 255) / 256, 256, 0, stream>>>(Pacc, nzero);

    pad_w_kernel<<<(NDIM * CP + 255) / 256, 256, 0, stream>>>(W, Wp);

    // big GEMM: px = x @ Wp  (TDM double-buffered path when available)
#if USE_TDM
    gemm_px_tdm_kernel<<<BATCH / 16, 256, 0, stream>>>(x, Wp, px);
#else
    gemm_wmma_kernel<<<BATCH / 16, 256, 0, stream>>>(x, Wp, px, BATCH, NDIM,
                                                     (long)NDIM, 1L);
#endif
    // tiny GEMMs (guarded generic kernel)
    gemm_wmma_kernel<<<(NCLS + 15) / 16, 256, 0, stream>>>(anchors, Wp, pa, NCLS, NDIM,
                                                           (long)NDIM, 1L);     // pa
    gemm_wmma_kernel<<<(NCLS + 15) / 16, 256, 0, stream>>>(W, Wp, G, NCLS, NDIM,
                                                           1L, (long)NCLS);     // G = W^T W

    pa_to_v_kernel<<<(NCLS + 7) / 8, 256, 0, stream>>>(pa, V);
    u2_kernel<<<NCLS, 128, 0, stream>>>(V, G, U2);
    sample_kernel<<<(BATCH + 7) / 8, 256, 0, stream>>>(px, y, U2, Pacc, counts, sums);
    final_kernel<<<1, 128, 0, stream>>>(Pacc, counts, U2, sums, out);
}